// TransLoss_58351425683828
// MI455X (gfx1250) — compile-verified
//
#include <hip/hip_runtime.h>
#include <hip/hip_bf16.h>
#include <stdint.h>

#define NEG_INF (-1e30f)

// ---------------------------------------------------------------------------
// Phase 1: per-row log-softmax statistics.
// rows = B*T*(U+1), each row has V=1024 fp32 logits (contiguous).
// One block = 8 rows (32 KB) staged into LDS via gfx1250 async global->LDS
// copies (ASYNCcnt path), then one wave32 per row does max / sum-exp shuffle
// reductions and emits blank_lp[row] and label_lp[row].
// ---------------------------------------------------------------------------
__global__ __launch_bounds__(256) void rnnt_rowstats(
    const float* __restrict__ pred, const int* __restrict__ target,
    float* __restrict__ blank_out, float* __restrict__ label_out,
    int T, int U, int V /* == 1024 */)
{
  __shared__ __align__(16) float lds[8 * 1024];
  const int tid  = threadIdx.x;
  const int wave = tid >> 5;
  const int lane = tid & 31;

  // ---- async copy 8 contiguous rows (32 KB) into LDS ----------------------
  // GVS mode: saddr = 64-bit base SGPR pair, vaddr = 32-bit byte offset,
  // vdst = LDS byte address. Low 32 bits of a generic LDS pointer are the
  // workgroup-relative LDS byte offset (ISA 10.2 aperture layout).
  uint32_t ldsbase = (uint32_t)(uintptr_t)(&lds[0]);
  const char* gbase = (const char*)(pred + (size_t)blockIdx.x * 8u * 1024u);
#pragma unroll
  for (int i = 0; i < 8; ++i) {
    uint32_t off = (uint32_t)(tid + i * 256) * 16u;   // 16B chunk per lane
    uint32_t la  = ldsbase + off;
    asm volatile("global_load_async_to_lds_b128 %0, %1, %2"
                 :: "v"(la), "v"(off), "s"(gbase) : "memory");
  }
  asm volatile("s_wait_asynccnt 0" ::: "memory");
  __syncthreads();   // every wave waited its own ASYNCcnt -> all 32 KB valid

  // ---- wave-per-row reduction over 1024 floats ----------------------------
  const float* row = &lds[wave * 1024];
  float m = NEG_INF;
#pragma unroll
  for (int i = 0; i < 8; ++i) {
    float4 v = *(const float4*)&row[i * 128 + lane * 4];
    m = fmaxf(m, fmaxf(fmaxf(v.x, v.y), fmaxf(v.z, v.w)));
  }
#pragma unroll
  for (int o = 16; o > 0; o >>= 1) m = fmaxf(m, __shfl_xor(m, o, 32));

  float s = 0.f;
#pragma unroll
  for (int i = 0; i < 8; ++i) {
    float4 v = *(const float4*)&row[i * 128 + lane * 4];
    s += __expf(v.x - m) + __expf(v.y - m) + __expf(v.z - m) + __expf(v.w - m);
  }
#pragma unroll
  for (int o = 16; o > 0; o >>= 1) s += __shfl_xor(s, o, 32);

  if (lane == 0) {
    float lsz  = m + __logf(s);          // logsumexp of the row
    int rowid  = blockIdx.x * 8 + wave;
    int U1     = U + 1;
    int b      = rowid / (T * U1);
    int rem    = rowid % (T * U1);
    int u      = rem % U1;
    blank_out[rowid] = row[0] - lsz;     // log P(blank | t,u)
    float lab = NEG_INF;                 // padded column u == U
    if (u < U) lab = row[target[b * U + u]] - lsz;
    label_out[rowid] = lab;
  }
}

// ---------------------------------------------------------------------------
// Phase 2: forward alphas via anti-diagonal wavefront, one block per batch.
// alpha[t][u] = logaddexp(alpha[t-1][u] + blank[t-1][u],
//                         alpha[t][u-1] + label[t][u-1])
// Everything (blank, label, alpha) lives in LDS (~100 KB, fine on CDNA5's
// 320 KB WGP LDS). 192 diagonal steps, one thread per column u.
// ---------------------------------------------------------------------------
__global__ __launch_bounds__(96) void rnnt_alpha(
    const float* __restrict__ blank, const float* __restrict__ label,
    const int* __restrict__ pred_len, const int* __restrict__ target_len,
    float* __restrict__ ll)
{
  constexpr int T  = 128;
  constexpr int U  = 64;
  constexpr int U1 = U + 1;
  constexpr int TU = T * U1;             // 8320
  __shared__ float sB[TU];
  __shared__ float sL[TU];
  __shared__ float sA[TU];

  const int b = blockIdx.x;
  for (int i = threadIdx.x; i < TU; i += blockDim.x) {
    sB[i] = blank[b * TU + i];
    sL[i] = label[b * TU + i];
  }
  __syncthreads();

  const int u = threadIdx.x;             // threads 65..95 idle in compute
  for (int d = 0; d < T + U; ++d) {      // diagonals 0..191
    if (u <= U) {
      int t = d - u;
      if (t >= 0 && t < T) {
        float v;
        if (t == 0) {
          v = (u == 0) ? 0.f : sA[u - 1] + sL[u - 1];          // cumsum row 0
        } else if (u == 0) {
          v = sA[(t - 1) * U1] + sB[(t - 1) * U1];             // blank column
        } else {
          float x = sA[(t - 1) * U1 + u] + sB[(t - 1) * U1 + u];
          float y = sA[t * U1 + (u - 1)] + sL[t * U1 + (u - 1)];
          float mx = fmaxf(x, y), mn = fminf(x, y);
          v = mx + __logf(1.f + __expf(mn - mx));              // logaddexp
        }
        sA[t * U1 + u] = v;
      }
    }
    __syncthreads();
  }

  if (threadIdx.x == 0) {
    int ti = pred_len[b] - 1;
    int ui = target_len[b];
    ll[b] = sA[ti * U1 + ui] + sB[ti * U1 + ui];
  }
}

// ---------------------------------------------------------------------------
// Phase 3: loss = -mean(ll)
// ---------------------------------------------------------------------------
__global__ void rnnt_finalize(const float* __restrict__ ll,
                              float* __restrict__ out, int B)
{
  if (threadIdx.x == 0 && blockIdx.x == 0) {
    float s = 0.f;
    for (int i = 0; i < B; ++i) s += ll[i];
    out[0] = -s / (float)B;
  }
}

// ---------------------------------------------------------------------------
extern "C" void kernel_launch(void* const* d_in, const int* in_sizes, int n_in,
                              void* d_out, int out_size, void* d_ws, size_t ws_size,
                              hipStream_t stream) {
  const float* pred       = (const float*)d_in[0];  // [B, T, U+1, V] fp32
  const int*   target     = (const int*)d_in[1];    // [B, U]
  const int*   pred_len   = (const int*)d_in[2];    // [B]
  const int*   target_len = (const int*)d_in[3];    // [B]

  const int B  = in_sizes[2];                       // 8
  const int U  = in_sizes[1] / B;                   // 64
  const int U1 = U + 1;
  const int V  = 1024;
  const int T  = (int)((long long)in_sizes[0] / ((long long)B * U1 * V)); // 128

  const int nrows = B * T * U1;                     // 66560

  float* blank = (float*)d_ws;                      // [nrows]
  float* label = blank + nrows;                     // [nrows]
  float* ll    = label + nrows;                     // [B]

  rnnt_rowstats<<<nrows / 8, 256, 0, stream>>>(pred, target, blank, label, T, U, V);
  rnnt_alpha<<<B, 96, 0, stream>>>(blank, label, pred_len, target_len, ll);
  rnnt_finalize<<<1, 32, 0, stream>>>(ll, (float*)d_out, B);
}